// CAF_71854802862201
// MI455X (gfx1250) — compile-verified
//
#include <hip/hip_runtime.h>

typedef __attribute__((ext_vector_type(2))) float v2f;
typedef __attribute__((ext_vector_type(8))) float v8f;

#define NNODES 100000
#define NEDGES 600000
#define DIM    128
#define BN_EPS 1e-5f
#define L2_EPS 1e-12f

// ---------------------------------------------------------------------------
// Edge scatter, one wave32 per edge: each lane gathers a float4 (512B
// coalesced per edge -> global_load_b128) and issues 4 f32 atomics into agg.
// Whole working set (x, agg, h ~ 154MB) is L2-resident on MI455X (192MB L2).
// ---------------------------------------------------------------------------
__global__ __launch_bounds__(256)
void sage_scatter1(const float* __restrict__ x,
                   const int* __restrict__ src,
                   const int* __restrict__ dst,
                   float* __restrict__ agg,
                   float* __restrict__ cnt) {
  long long t = (long long)blockIdx.x * 256 + threadIdx.x;
  int e    = (int)(t >> 5);
  int lane = (int)(t & 31);
  if (e >= NEDGES) return;
  int s = src[e];
  int d = dst[e];
  float4 v = ((const float4*)(x + (size_t)s * DIM))[lane];
  float* ap = agg + (size_t)d * DIM + lane * 4;
  atomicAdd(ap + 0, v.x);
  atomicAdd(ap + 1, v.y);
  atomicAdd(ap + 2, v.z);
  atomicAdd(ap + 3, v.w);
  if (lane == 0) atomicAdd(&cnt[d], 1.0f);
}

// Layer-2 scatter with BatchNorm affine applied to the message on the fly.
__global__ __launch_bounds__(256)
void sage_scatter2(const float* __restrict__ h,
                   const int* __restrict__ src,
                   const int* __restrict__ dst,
                   const float* __restrict__ scale,
                   const float* __restrict__ shift,
                   float* __restrict__ agg) {
  long long t = (long long)blockIdx.x * 256 + threadIdx.x;
  int e    = (int)(t >> 5);
  int lane = (int)(t & 31);
  if (e >= NEDGES) return;
  int s = src[e];
  int d = dst[e];
  float4 v  = ((const float4*)(h + (size_t)s * DIM))[lane];
  float4 sc = ((const float4*)scale)[lane];
  float4 sh = ((const float4*)shift)[lane];
  v.x = v.x * sc.x + sh.x;
  v.y = v.y * sc.y + sh.y;
  v.z = v.z * sc.z + sh.z;
  v.w = v.w * sc.w + sh.w;
  float* ap = agg + (size_t)d * DIM + lane * 4;
  atomicAdd(ap + 0, v.x);
  atomicAdd(ap + 1, v.y);
  atomicAdd(ap + 2, v.z);
  atomicAdd(ap + 3, v.w);
}

// Row-group L2-norm helper: 16 threads per row, width-16 xor-shuffle tree
// (wave32-safe: each 16-lane group stays inside one wave half).
__device__ __forceinline__ float rowgroup_reduce16(float v) {
  v += __shfl_xor(v, 1, 16);
  v += __shfl_xor(v, 2, 16);
  v += __shfl_xor(v, 4, 16);
  v += __shfl_xor(v, 8, 16);
  return v;
}

// ---------------------------------------------------------------------------
// Fused SAGE layer 1:
//   out = mean @ W_l + b_l + x @ W_r   -> row-L2-normalize -> ReLU -> h
//   + per-column BN sum / sum-of-squares accumulation.
// One block = 16 rows; 8 waves each own a 16-column slice; f32 WMMA 16x16x4.
// ---------------------------------------------------------------------------
__global__ __launch_bounds__(256)
void sage_gemm1(const float* __restrict__ x,
                const float* __restrict__ agg,
                const float* __restrict__ cnt,
                const float* __restrict__ Wl,
                const float* __restrict__ bl,
                const float* __restrict__ Wr,
                float* __restrict__ hout,
                float* __restrict__ bnsum,
                float* __restrict__ bnsq) {
  __shared__ alignas(16) float xls[16][DIM];
  __shared__ alignas(16) float mls[16][DIM];
  __shared__ alignas(16) float outls[16][DIM];
  __shared__ float rcnt[16];
  __shared__ float rinv[16];
  __shared__ float csum[DIM];
  __shared__ float csq[DIM];

  const int tid = threadIdx.x;
  const int r0  = blockIdx.x * 16;        // N % 16 == 0, no masking needed

  if (tid < 16)  rcnt[tid] = 1.0f / fmaxf(cnt[r0 + tid], 1.0f);
  if (tid < DIM) { csum[tid] = 0.0f; csq[tid] = 0.0f; }
  // stage x tile (float4 -> global_load_b128 + ds_store_b128)
  for (int i = tid; i < 16 * 32; i += 256) {
    int r = i >> 5, c4 = i & 31;
    ((float4*)xls[r])[c4] = ((const float4*)(x + (size_t)(r0 + r) * DIM))[c4];
  }
  __syncthreads();
  // stage mean tile = agg / max(cnt,1)
  for (int i = tid; i < 16 * 32; i += 256) {
    int r = i >> 5, c4 = i & 31;
    float4 v = ((const float4*)(agg + (size_t)(r0 + r) * DIM))[c4];
    float rc = rcnt[r];
    v.x *= rc; v.y *= rc; v.z *= rc; v.w *= rc;
    ((float4*)mls[r])[c4] = v;
  }
  __syncthreads();

  const int wave = tid >> 5;
  const int lane = tid & 31;
  const int half = lane >> 4;             // selects K pair within the 16x4 frag
  const int l16  = lane & 15;             // M index (A) / N index (B)
  const int c0   = wave * 16;

  v8f acc = {};
  #pragma unroll 4
  for (int k0 = 0; k0 < DIM; k0 += 4) {
    int k = k0 + 2 * half;
    v2f am, ax, bwl, bwr;
    am.x  = mls[l16][k];            am.y  = mls[l16][k + 1];
    ax.x  = xls[l16][k];            ax.y  = xls[l16][k + 1];
    bwl.x = Wl[k * DIM + c0 + l16]; bwl.y = Wl[(k + 1) * DIM + c0 + l16];
    bwr.x = Wr[k * DIM + c0 + l16]; bwr.y = Wr[(k + 1) * DIM + c0 + l16];
    acc = __builtin_amdgcn_wmma_f32_16x16x4_f32(false, am, false, bwl,
                                                (short)0, acc, false, false);
    acc = __builtin_amdgcn_wmma_f32_16x16x4_f32(false, ax, false, bwr,
                                                (short)0, acc, false, false);
  }

  float bias = bl[c0 + l16];
  #pragma unroll
  for (int v = 0; v < 8; ++v)
    outls[v + 8 * half][c0 + l16] = acc[v] + bias;
  __syncthreads();

  // parallel row L2-norm: 16 threads per row, shuffle-reduced
  {
    int r  = tid >> 4;
    int cl = tid & 15;
    float s = 0.0f;
    #pragma unroll
    for (int j = 0; j < 8; ++j) {
      float v = outls[r][cl + 16 * j];
      s += v * v;
    }
    s = rowgroup_reduce16(s);
    if (cl == 0) rinv[r] = 1.0f / fmaxf(sqrtf(s), L2_EPS);
  }
  __syncthreads();

  // normalize + ReLU + store h (float4) + BN partial sums
  {
    int c4 = tid & 31;               // float4 column
    float ls0 = 0, ls1 = 0, ls2 = 0, ls3 = 0;
    float lq0 = 0, lq1 = 0, lq2 = 0, lq3 = 0;
    for (int r = (tid >> 5); r < 16; r += 8) {
      float ri = rinv[r];
      float4 v = ((const float4*)outls[r])[c4];
      v.x = fmaxf(v.x * ri, 0.0f);
      v.y = fmaxf(v.y * ri, 0.0f);
      v.z = fmaxf(v.z * ri, 0.0f);
      v.w = fmaxf(v.w * ri, 0.0f);
      ((float4*)(hout + (size_t)(r0 + r) * DIM))[c4] = v;
      ls0 += v.x; ls1 += v.y; ls2 += v.z; ls3 += v.w;
      lq0 += v.x * v.x; lq1 += v.y * v.y; lq2 += v.z * v.z; lq3 += v.w * v.w;
    }
    int c = c4 * 4;
    atomicAdd(&csum[c + 0], ls0); atomicAdd(&csq[c + 0], lq0);
    atomicAdd(&csum[c + 1], ls1); atomicAdd(&csq[c + 1], lq1);
    atomicAdd(&csum[c + 2], ls2); atomicAdd(&csq[c + 2], lq2);
    atomicAdd(&csum[c + 3], ls3); atomicAdd(&csq[c + 3], lq3);
  }
  __syncthreads();
  if (tid < DIM) {
    atomicAdd(&bnsum[tid], csum[tid]);
    atomicAdd(&bnsq[tid],  csq[tid]);
  }
}

// Fold BN batch statistics into per-column scale/shift.
__global__ void bn_finalize(const float* __restrict__ sums,
                            const float* __restrict__ sumsq,
                            const float* __restrict__ gamma,
                            const float* __restrict__ beta,
                            float* __restrict__ scale,
                            float* __restrict__ shift) {
  int c = threadIdx.x;
  float inv_n = 1.0f / (float)NNODES;
  float mu  = sums[c] * inv_n;
  float var = sumsq[c] * inv_n - mu * mu;
  float sc  = gamma[c] * rsqrtf(var + BN_EPS);
  scale[c] = sc;
  shift[c] = beta[c] - mu * sc;
}

// ---------------------------------------------------------------------------
// Fused SAGE layer 2 + fc head:
//   xbn = h*scale+shift ; out = (agg/cnt) @ W_l + b_l + xbn @ W_r
//   -> row-L2-normalize -> embed ; preds = embed[:, :64] @ fcW + fcb
// ---------------------------------------------------------------------------
__global__ __launch_bounds__(256)
void sage_gemm2(const float* __restrict__ h,
                const float* __restrict__ agg,
                const float* __restrict__ cnt,
                const float* __restrict__ scale,
                const float* __restrict__ shift,
                const float* __restrict__ Wl,
                const float* __restrict__ bl,
                const float* __restrict__ Wr,
                const float* __restrict__ fcW,
                const float* __restrict__ fcb,
                float* __restrict__ preds,
                float* __restrict__ embed) {
  __shared__ alignas(16) float xls[16][DIM];
  __shared__ alignas(16) float mls[16][DIM];
  __shared__ alignas(16) float outls[16][DIM];
  __shared__ float rcnt[16];
  __shared__ float rinv[16];

  const int tid = threadIdx.x;
  const int r0  = blockIdx.x * 16;

  if (tid < 16) rcnt[tid] = 1.0f / fmaxf(cnt[r0 + tid], 1.0f);
  // stage BN(h) tile
  for (int i = tid; i < 16 * 32; i += 256) {
    int r = i >> 5, c4 = i & 31;
    float4 v  = ((const float4*)(h + (size_t)(r0 + r) * DIM))[c4];
    float4 sc = ((const float4*)scale)[c4];
    float4 sh = ((const float4*)shift)[c4];
    v.x = v.x * sc.x + sh.x;
    v.y = v.y * sc.y + sh.y;
    v.z = v.z * sc.z + sh.z;
    v.w = v.w * sc.w + sh.w;
    ((float4*)xls[r])[c4] = v;
  }
  __syncthreads();
  for (int i = tid; i < 16 * 32; i += 256) {
    int r = i >> 5, c4 = i & 31;
    float4 v = ((const float4*)(agg + (size_t)(r0 + r) * DIM))[c4];
    float rc = rcnt[r];
    v.x *= rc; v.y *= rc; v.z *= rc; v.w *= rc;
    ((float4*)mls[r])[c4] = v;
  }
  __syncthreads();

  const int wave = tid >> 5;
  const int lane = tid & 31;
  const int half = lane >> 4;
  const int l16  = lane & 15;
  const int c0   = wave * 16;

  v8f acc = {};
  #pragma unroll 4
  for (int k0 = 0; k0 < DIM; k0 += 4) {
    int k = k0 + 2 * half;
    v2f am, ax, bwl, bwr;
    am.x  = mls[l16][k];            am.y  = mls[l16][k + 1];
    ax.x  = xls[l16][k];            ax.y  = xls[l16][k + 1];
    bwl.x = Wl[k * DIM + c0 + l16]; bwl.y = Wl[(k + 1) * DIM + c0 + l16];
    bwr.x = Wr[k * DIM + c0 + l16]; bwr.y = Wr[(k + 1) * DIM + c0 + l16];
    acc = __builtin_amdgcn_wmma_f32_16x16x4_f32(false, am, false, bwl,
                                                (short)0, acc, false, false);
    acc = __builtin_amdgcn_wmma_f32_16x16x4_f32(false, ax, false, bwr,
                                                (short)0, acc, false, false);
  }

  float bias = bl[c0 + l16];
  #pragma unroll
  for (int v = 0; v < 8; ++v)
    outls[v + 8 * half][c0 + l16] = acc[v] + bias;
  __syncthreads();

  {
    int r  = tid >> 4;
    int cl = tid & 15;
    float s = 0.0f;
    #pragma unroll
    for (int j = 0; j < 8; ++j) {
      float v = outls[r][cl + 16 * j];
      s += v * v;
    }
    s = rowgroup_reduce16(s);
    if (cl == 0) rinv[r] = 1.0f / fmaxf(sqrtf(s), L2_EPS);
  }
  __syncthreads();

  // normalize, write embed (float4), keep normalized copy in LDS for fc head
  {
    int c4 = tid & 31;
    for (int r = (tid >> 5); r < 16; r += 8) {
      float ri = rinv[r];
      float4 v = ((const float4*)outls[r])[c4];
      v.x *= ri; v.y *= ri; v.z *= ri; v.w *= ri;
      ((float4*)outls[r])[c4] = v;
      ((float4*)(embed + (size_t)(r0 + r) * DIM))[c4] = v;
    }
  }
  __syncthreads();

  // fc head on causal half: preds[row] = embed[row, :64] @ fcW + fcb
  if (tid < 16) {
    float a = fcb[0];
    for (int c = 0; c < DIM / 2; ++c) a += outls[tid][c] * fcW[c];
    preds[r0 + tid] = a;
  }
}

// ---------------------------------------------------------------------------
extern "C" void kernel_launch(void* const* d_in, const int* in_sizes, int n_in,
                              void* d_out, int out_size, void* d_ws, size_t ws_size,
                              hipStream_t stream) {
  (void)in_sizes; (void)n_in; (void)out_size; (void)ws_size;

  const float* x   = (const float*)d_in[0];
  const int*   ei  = (const int*)d_in[1];
  const float* W1l = (const float*)d_in[2];
  const float* b1l = (const float*)d_in[3];
  const float* W1r = (const float*)d_in[4];
  const float* gam = (const float*)d_in[5];
  const float* bet = (const float*)d_in[6];
  const float* W2l = (const float*)d_in[7];
  const float* b2l = (const float*)d_in[8];
  const float* W2r = (const float*)d_in[9];
  const float* fcW = (const float*)d_in[10];
  const float* fcb = (const float*)d_in[11];

  const int* src = ei;            // edge_index row 0
  const int* dst = ei + NEDGES;   // edge_index row 1

  float* ws      = (float*)d_ws;
  float* agg     = ws;                             // N*128
  float* cnt     = agg + (size_t)NNODES * DIM;     // N
  float* hbuf    = cnt + NNODES;                   // N*128
  float* bnsum   = hbuf + (size_t)NNODES * DIM;    // 128
  float* bnsq    = bnsum + DIM;                    // 128
  float* bnscale = bnsq + DIM;                     // 128
  float* bnshift = bnscale + DIM;                  // 128

  float* preds = (float*)d_out;       // [N]
  float* embed = preds + NNODES;      // [N,128]

  const int sgrid = (NEDGES * 32 + 255) / 256;     // one wave32 per edge
  const int ggrid = NNODES / 16;                   // 6250 blocks

  hipMemsetAsync(agg, 0, (size_t)NNODES * DIM * sizeof(float), stream);
  hipMemsetAsync(cnt, 0, (size_t)NNODES * sizeof(float), stream);
  hipMemsetAsync(bnsum, 0, 2 * DIM * sizeof(float), stream);

  sage_scatter1<<<sgrid, 256, 0, stream>>>(x, src, dst, agg, cnt);
  sage_gemm1<<<ggrid, 256, 0, stream>>>(x, agg, cnt, W1l, b1l, W1r,
                                        hbuf, bnsum, bnsq);
  bn_finalize<<<1, DIM, 0, stream>>>(bnsum, bnsq, gam, bet, bnscale, bnshift);

  hipMemsetAsync(agg, 0, (size_t)NNODES * DIM * sizeof(float), stream);
  sage_scatter2<<<sgrid, 256, 0, stream>>>(hbuf, src, dst, bnscale, bnshift, agg);
  sage_gemm2<<<ggrid, 256, 0, stream>>>(hbuf, agg, cnt, bnscale, bnshift,
                                        W2l, b2l, W2r, fcW, fcb, preds, embed);
}